// VariationalLinear_19774029431036
// MI455X (gfx1250) — compile-verified
//
#include <hip/hip_runtime.h>
#include <hip/hip_bf16.h>

typedef __attribute__((ext_vector_type(2))) float v2f;
typedef __attribute__((ext_vector_type(4))) float v4f;
typedef __attribute__((ext_vector_type(8))) float v8f;

#define IN_F   4096
#define OUT_F  4096
#define BATCH  256
#define KC     32      // K chunk per iteration
#define NT     32      // output features per workgroup
#define LSTR   34      // LDS row stride (floats): even (8B-aligned float2) + bank-conflict free

#define LOG2E_F 1.4426950408889634f
#define LN2_F   0.6931471805599453f

// Raw hardware transcendentals (v_exp_f32 / v_log_f32 are base-2; inputs here
// are always normal-range so no denormal fixups needed).
__device__ __forceinline__ float exp2_raw(float x) { return __builtin_amdgcn_exp2f(x); }
__device__ __forceinline__ float log2_raw(float x) { return __builtin_amdgcn_logf(x); }
__device__ __forceinline__ float rcp_raw(float x)  { return __builtin_amdgcn_rcpf(x); }

// softplus(x) = max(x,0) + ln(1 + 2^(-|x|*log2e))
__device__ __forceinline__ float softplus_f(float x) {
    const float t = exp2_raw(-fabsf(x) * LOG2E_F);
    return fmaxf(x, 0.0f) + log2_raw(1.0f + t) * LN2_F;
}

__global__ void kl_zero_kernel(float* kl) { *kl = 0.0f; }

__global__ __launch_bounds__(256) void vlin_kernel(
    const float* __restrict__ x,     // [BATCH, IN_F]
    const float* __restrict__ wmu,   // [OUT_F, IN_F]
    const float* __restrict__ wrho,  // [OUT_F, IN_F]
    const float* __restrict__ bmu,   // [OUT_F]
    const float* __restrict__ brho,  // [OUT_F]
    const float* __restrict__ epsw,  // [OUT_F, IN_F]
    const float* __restrict__ epsb,  // [OUT_F]
    const float* __restrict__ pmu,   // [OUT_F, IN_F]
    const float* __restrict__ psig,  // [OUT_F, IN_F]
    float* __restrict__ out,         // [BATCH, OUT_F]
    float* __restrict__ kl_out)      // scalar
{
    __shared__ __align__(16) float lds_x[BATCH * LSTR]; // 34816 B
    __shared__ __align__(16) float lds_w[NT * LSTR];    //  4352 B
    __shared__ float red[256];

    const unsigned tid  = threadIdx.x;
    const unsigned lane = tid & 31u;
    const unsigned wv   = tid >> 5;          // wave 0..7
    const unsigned half = lane >> 4;         // 0/1
    const unsigned l16  = lane & 15u;
    const unsigned o0   = blockIdx.x * NT;

    const unsigned lrow = tid >> 3;          // 0..31
    const unsigned lk4  = (tid & 7u) * 4u;   // 0,4,...,28

    const v8f vzero = {0.f,0.f,0.f,0.f,0.f,0.f,0.f,0.f};
    v8f acc[2][2];
    acc[0][0] = vzero; acc[0][1] = vzero; acc[1][0] = vzero; acc[1][1] = vzero;

    float kl_lin = 0.0f;   // quadratic terms
    float kl_l2  = 0.0f;   // log2-domain terms (scaled by ln2 at the end)

    for (unsigned k0 = 0; k0 < IN_F; k0 += KC) {
        // ---- stage x tile: 256 rows x 32 k (coalesced 16B loads) ----
        #pragma unroll
        for (unsigned it = 0; it < 8; ++it) {
            const unsigned m = it * 32u + lrow;
            const v4f xv = *reinterpret_cast<const v4f*>(&x[(size_t)m * IN_F + k0 + lk4]);
            float* dst = &lds_x[m * LSTR + lk4];
            v2f p0; p0.x = xv.x; p0.y = xv.y;
            v2f p1; p1.x = xv.z; p1.y = xv.w;
            *reinterpret_cast<v2f*>(dst)     = p0;
            *reinterpret_cast<v2f*>(dst + 2) = p1;
        }

        // ---- stage W tile: weight = mu + softplus(rho)*eps; accumulate KL ----
        {
            const unsigned o = lrow;  // 0..31
            const size_t gi = (size_t)(o0 + o) * IN_F + k0 + lk4;
            // streamed once device-wide -> non-temporal (keep L2 for x reuse)
            const v4f mu = __builtin_nontemporal_load(reinterpret_cast<const v4f*>(&wmu[gi]));
            const v4f rh = __builtin_nontemporal_load(reinterpret_cast<const v4f*>(&wrho[gi]));
            const v4f ew = __builtin_nontemporal_load(reinterpret_cast<const v4f*>(&epsw[gi]));
            const v4f pm = __builtin_nontemporal_load(reinterpret_cast<const v4f*>(&pmu[gi]));
            const v4f ps = __builtin_nontemporal_load(reinterpret_cast<const v4f*>(&psig[gi]));
            v4f w;
            #pragma unroll
            for (int j = 0; j < 4; ++j) {
                const float sg = softplus_f(rh[j]);
                w[j] = fmaf(sg, ew[j], mu[j]);
                const float dm   = mu[j] - pm[j];
                const float inv2 = 0.5f * rcp_raw(ps[j] * ps[j]);
                kl_l2  += log2_raw(ps[j]) - log2_raw(sg);
                kl_lin += (sg * sg + dm * dm) * inv2;
            }
            float* dst = &lds_w[o * LSTR + lk4];
            v2f q0; q0.x = w.x; q0.y = w.y;
            v2f q1; q1.x = w.z; q1.y = w.w;
            *reinterpret_cast<v2f*>(dst)     = q0;
            *reinterpret_cast<v2f*>(dst + 2) = q1;
        }
        __syncthreads();

        // ---- WMMA: each wave computes its 32x32 block as 2x2 16x16 tiles ----
        const unsigned mbase = wv * 32u + l16;
        #pragma unroll
        for (unsigned kk = 0; kk < KC; kk += 4) {
            const unsigned ko = kk + 2u * half;  // A/B VGPR0/1 hold K = 2*half, 2*half+1
            const v2f a0 = *reinterpret_cast<const v2f*>(&lds_x[mbase          * LSTR + ko]);
            const v2f a1 = *reinterpret_cast<const v2f*>(&lds_x[(mbase + 16u)  * LSTR + ko]);
            const v2f b0 = *reinterpret_cast<const v2f*>(&lds_w[l16            * LSTR + ko]);
            const v2f b1 = *reinterpret_cast<const v2f*>(&lds_w[(l16 + 16u)    * LSTR + ko]);
            acc[0][0] = __builtin_amdgcn_wmma_f32_16x16x4_f32(false, a0, false, b0, (short)0, acc[0][0], false, false);
            acc[0][1] = __builtin_amdgcn_wmma_f32_16x16x4_f32(false, a0, false, b1, (short)0, acc[0][1], false, false);
            acc[1][0] = __builtin_amdgcn_wmma_f32_16x16x4_f32(false, a1, false, b0, (short)0, acc[1][0], false, false);
            acc[1][1] = __builtin_amdgcn_wmma_f32_16x16x4_f32(false, a1, false, b1, (short)0, acc[1][1], false, false);
        }
        __syncthreads();
    }

    // ---- epilogue: fused bias + streamed store ----
    #pragma unroll
    for (unsigned ni = 0; ni < 2; ++ni) {
        const unsigned o = o0 + ni * 16u + l16;
        const float bias = fmaf(softplus_f(brho[o]), epsb[o], bmu[o]);
        #pragma unroll
        for (unsigned mi = 0; mi < 2; ++mi) {
            const unsigned mrow = wv * 32u + mi * 16u + 8u * half; // C/D: VGPR v -> row v (+8 hi half)
            #pragma unroll
            for (int v = 0; v < 8; ++v) {
                __builtin_nontemporal_store(acc[mi][ni][v] + bias,
                                            &out[(size_t)(mrow + v) * OUT_F + o]);
            }
        }
    }

    // ---- KL: combine domains; each thread summed 4*128 = 512 elements (-0.5 each) ----
    float kl = fmaf(kl_l2, LN2_F, kl_lin) - 256.0f;

    red[tid] = kl;
    __syncthreads();
    #pragma unroll
    for (unsigned s = 128; s > 0; s >>= 1) {
        if (tid < s) red[tid] += red[tid + s];
        __syncthreads();
    }
    if (tid == 0) atomicAdd(kl_out, red[0]);
}

extern "C" void kernel_launch(void* const* d_in, const int* in_sizes, int n_in,
                              void* d_out, int out_size, void* d_ws, size_t ws_size,
                              hipStream_t stream) {
    (void)in_sizes; (void)n_in; (void)out_size; (void)d_ws; (void)ws_size;
    const float* x    = (const float*)d_in[0];
    const float* wmu  = (const float*)d_in[1];
    const float* wrho = (const float*)d_in[2];
    const float* bmu  = (const float*)d_in[3];
    const float* brho = (const float*)d_in[4];
    const float* epsw = (const float*)d_in[5];
    const float* epsb = (const float*)d_in[6];
    const float* pmu  = (const float*)d_in[7];
    const float* psig = (const float*)d_in[8];

    float* out = (float*)d_out;
    float* kl  = out + (size_t)BATCH * OUT_F;

    kl_zero_kernel<<<1, 1, 0, stream>>>(kl);
    vlin_kernel<<<OUT_F / NT, 256, 0, stream>>>(x, wmu, wrho, bmu, brho,
                                                epsw, epsb, pmu, psig, out, kl);
}